// GATLayer_18648747999679
// MI455X (gfx1250) — compile-verified
//
#include <hip/hip_runtime.h>

// Problem constants (match reference)
#define M_NODES 100000
#define N_NODES 50000
#define DIM     128
#define NEDGE   1000000
#define NEG_SLOPE 0.2f

typedef __attribute__((ext_vector_type(2))) float v2f;
typedef __attribute__((ext_vector_type(8))) float v8f;

// ---------------------------------------------------------------------------
// float atomic max via int/uint ordering trick (lowers to global_atomic_max_i32
// / global_atomic_min_u32 — native CDNA5 atomics, correct for all sign combos)
// ---------------------------------------------------------------------------
__device__ __forceinline__ void atomicMaxF(float* addr, float val) {
    if (val >= 0.0f) {
        atomicMax((int*)addr, __float_as_int(val));
    } else {
        atomicMin((unsigned int*)addr, __float_as_uint(val));
    }
}

__device__ __forceinline__ float leaky(float x) {
    return x > 0.0f ? x : NEG_SLOPE * x;
}

// ---------------------------------------------------------------------------
// Fill kernel (grid-stride). Re-inits output/max/sum buffers every call so the
// atomic accumulation is deterministic across graph replays.
// ---------------------------------------------------------------------------
__global__ __launch_bounds__(256) void fill_kernel(float* __restrict__ p, float v, size_t n) {
    size_t i = (size_t)blockIdx.x * blockDim.x + threadIdx.x;
    size_t stride = (size_t)gridDim.x * blockDim.x;
    for (; i < n; i += stride) p[i] = v;
}

// ---------------------------------------------------------------------------
// Full-precision WMMA GEMM: out[rows x 128] = A[rows x 128] @ W[128 x 128] + b
// One wave computes a 16-row x 128-col stripe: 8 accumulator tiles (v8f each),
// K swept in 32 steps of V_WMMA_F32_16X16X4_F32. W staged once in 64 KB LDS.
// The A fragment is software-pipelined by one k-step so the global_load for
// step kk+1 is in flight while the 8 WMMAs of step kk execute (removes the
// load->s_wait_loadcnt 0->wmma stall seen in the previous round's asm).
//
// f32 WMMA VGPR layout (ISA 7.12.2):
//   A 16x4 : lane L<16 -> row M=L, {K0,K1}; lane L>=16 -> row M=L-16, {K2,K3}
//   B 4x16 : lane L<16 -> col N=L, {K0,K1}; lane L>=16 -> col N=L-16, {K2,K3}
//   C 16x16: vgpr r, lanes 0-15 -> (M=r,   N=lane)
//                     lanes16-31 -> (M=r+8, N=lane-16)
// ---------------------------------------------------------------------------
__global__ __launch_bounds__(256) void gemm_wmma_f32(
    const float* __restrict__ A, const float* __restrict__ W,
    const float* __restrict__ bias, float* __restrict__ out, int rows)
{
    __shared__ float wlds[DIM * DIM];   // 64 KB
    for (int t = threadIdx.x; t < DIM * DIM; t += 256) wlds[t] = W[t];
    __syncthreads();

    const int wave = threadIdx.x >> 5;
    const int lane = threadIdx.x & 31;
    const int tile = blockIdx.x * 8 + wave;
    if (tile * 16 >= rows) return;          // rows is a multiple of 16

    const int row0 = tile * 16;
    const int mrow = lane & 15;             // row-in-tile (A) / col-in-tile (B)
    const int koff = (lane >> 4) << 1;      // 0 for lanes 0-15, 2 for 16-31

    v8f acc[8] = {};                        // 8 N-tiles of 16x16 fp32

    const float* aRow = A + (size_t)(row0 + mrow) * DIM + koff;

    // Prime the pipeline with the kk=0 fragment.
    v2f a_cur;
    a_cur[0] = aRow[0];
    a_cur[1] = aRow[1];

    #pragma unroll 4
    for (int kk = 0; kk < 31; ++kk) {
        // Issue next A fragment load before consuming the current one.
        v2f a_nxt;
        a_nxt[0] = aRow[(kk + 1) * 4 + 0];
        a_nxt[1] = aRow[(kk + 1) * 4 + 1];

        const int bk = kk * 4 + koff;
        #pragma unroll
        for (int n = 0; n < 8; ++n) {
            v2f b;
            b[0] = wlds[bk * DIM + n * 16 + mrow];
            b[1] = wlds[(bk + 1) * DIM + n * 16 + mrow];
            acc[n] = __builtin_amdgcn_wmma_f32_16x16x4_f32(
                false, a_cur, false, b, (short)0, acc[n], false, false);
        }
        a_cur = a_nxt;
    }
    {   // Peeled last step (kk = 31): no over-read past the end of A.
        const int bk = 31 * 4 + koff;
        #pragma unroll
        for (int n = 0; n < 8; ++n) {
            v2f b;
            b[0] = wlds[bk * DIM + n * 16 + mrow];
            b[1] = wlds[(bk + 1) * DIM + n * 16 + mrow];
            acc[n] = __builtin_amdgcn_wmma_f32_16x16x4_f32(
                false, a_cur, false, b, (short)0, acc[n], false, false);
        }
    }

    const int rbase = row0 + ((lane >> 4) << 3);  // +8 rows for upper half-wave
    #pragma unroll
    for (int n = 0; n < 8; ++n) {
        const int colg = n * 16 + mrow;
        const float bv = bias[colg];
        #pragma unroll
        for (int r = 0; r < 8; ++r) {
            out[(size_t)(rbase + r) * DIM + colg] = acc[n][r] + bv;
        }
    }
}

// ---------------------------------------------------------------------------
// Per-node attention scalars: for each node m, 4 dot products of length 128:
//   sA[m,h] = feat[m] . aA[h]   (node as softmax-source in one graph)
//   sB[m,h] = feat[m] . aB[h]   (node as dest in the other graph)
// One wave32 per node; float4 loads + __shfl_xor reduction.
// ---------------------------------------------------------------------------
__global__ __launch_bounds__(256) void node_scores(
    const float* __restrict__ feat, int n,
    const float* __restrict__ aA, const float* __restrict__ aB,
    float* __restrict__ sA, float* __restrict__ sB)
{
    const int w = (int)(((size_t)blockIdx.x * blockDim.x + threadIdx.x) >> 5);
    const int lane = threadIdx.x & 31;
    if (w >= n) return;

    const float4 f = ((const float4*)(feat + (size_t)w * DIM))[lane];
    const float* av[4] = { aA, aA + DIM, aB, aB + DIM };
    float p[4];
    #pragma unroll
    for (int j = 0; j < 4; ++j) {
        const float4 a = ((const float4*)av[j])[lane];
        p[j] = f.x * a.x + f.y * a.y + f.z * a.z + f.w * a.w;
    }
    #pragma unroll
    for (int off = 16; off > 0; off >>= 1) {
        #pragma unroll
        for (int j = 0; j < 4; ++j) p[j] += __shfl_xor(p[j], off, 32);
    }
    if (lane == 0) {
        sA[w * 2 + 0] = p[0];  sA[w * 2 + 1] = p[1];
        sB[w * 2 + 0] = p[2];  sB[w * 2 + 1] = p[3];
    }
}

// ---------------------------------------------------------------------------
// Edge pass 1: segment max of logits per (row, head) via float atomic max.
// ---------------------------------------------------------------------------
__global__ __launch_bounds__(256) void edge_max(
    const int* __restrict__ row, const int* __restrict__ col,
    const float* __restrict__ ssrc, const float* __restrict__ sdst,
    float* __restrict__ mx, int ne)
{
    const int e = blockIdx.x * blockDim.x + threadIdx.x;
    if (e >= ne) return;
    const int r = row[e], c = col[e];
    atomicMaxF(&mx[r * 2 + 0], leaky(ssrc[r * 2 + 0] + sdst[c * 2 + 0]));
    atomicMaxF(&mx[r * 2 + 1], leaky(ssrc[r * 2 + 1] + sdst[c * 2 + 1]));
}

// ---------------------------------------------------------------------------
// Edge pass 2: segment sum of exp(logit - max) per (row, head).
// ---------------------------------------------------------------------------
__global__ __launch_bounds__(256) void edge_expsum(
    const int* __restrict__ row, const int* __restrict__ col,
    const float* __restrict__ ssrc, const float* __restrict__ sdst,
    const float* __restrict__ mx, float* __restrict__ sum, int ne)
{
    const int e = blockIdx.x * blockDim.x + threadIdx.x;
    if (e >= ne) return;
    const int r = row[e], c = col[e];
    #pragma unroll
    for (int h = 0; h < 2; ++h) {
        const float l = leaky(ssrc[r * 2 + h] + sdst[c * 2 + h]);
        atomicAdd(&sum[r * 2 + h], expf(l - mx[r * 2 + h]));
    }
}

// ---------------------------------------------------------------------------
// Edge pass 3: out[row] += mean_h(att_h) * h_dst[col].  Head-mean folds into a
// single scalar weight per edge since h_dst is head-independent. One wave per
// edge: float4 (b128) gather of the 128-dim neighbor row, 4 fp32 atomic adds
// per lane into the row-segment accumulator.
// ---------------------------------------------------------------------------
__global__ __launch_bounds__(256) void edge_scatter(
    const int* __restrict__ row, const int* __restrict__ col,
    const float* __restrict__ ssrc, const float* __restrict__ sdst,
    const float* __restrict__ mx, const float* __restrict__ sum,
    const float* __restrict__ hdst, float* __restrict__ acc, int ne)
{
    const int e = (int)(((size_t)blockIdx.x * blockDim.x + threadIdx.x) >> 5);
    const int lane = threadIdx.x & 31;
    if (e >= ne) return;
    const int r = row[e], c = col[e];

    float w = 0.0f;
    #pragma unroll
    for (int h = 0; h < 2; ++h) {
        const float l = leaky(ssrc[r * 2 + h] + sdst[c * 2 + h]);
        w += expf(l - mx[r * 2 + h]) / sum[r * 2 + h];
    }
    w *= 0.5f;

    const float4 hv = ((const float4*)(hdst + (size_t)c * DIM))[lane];
    float* ap = acc + (size_t)r * DIM + lane * 4;
    atomicAdd(ap + 0, w * hv.x);
    atomicAdd(ap + 1, w * hv.y);
    atomicAdd(ap + 2, w * hv.z);
    atomicAdd(ap + 3, w * hv.w);
}

// ---------------------------------------------------------------------------
// Final ELU in place on the accumulated output.
// ---------------------------------------------------------------------------
__global__ __launch_bounds__(256) void elu_kernel(float* __restrict__ p, size_t n) {
    size_t i = (size_t)blockIdx.x * blockDim.x + threadIdx.x;
    size_t stride = (size_t)gridDim.x * blockDim.x;
    for (; i < n; i += stride) {
        const float x = p[i];
        p[i] = x > 0.0f ? x : (expf(x) - 1.0f);
    }
}

// ---------------------------------------------------------------------------
extern "C" void kernel_launch(void* const* d_in, const int* in_sizes, int n_in,
                              void* d_out, int out_size, void* d_ws, size_t ws_size,
                              hipStream_t stream) {
    const float* u_prev  = (const float*)d_in[0];
    const float* i_prev  = (const float*)d_in[1];
    const float* w_user  = (const float*)d_in[2];
    const float* b_user  = (const float*)d_in[3];
    const float* w_item  = (const float*)d_in[4];
    const float* b_item  = (const float*)d_in[5];
    const float* a_u_src = (const float*)d_in[6];
    const float* a_u_dst = (const float*)d_in[7];
    const float* a_i_src = (const float*)d_in[8];
    const float* a_i_dst = (const float*)d_in[9];
    const int* u2i_row = (const int*)d_in[10];
    const int* u2i_col = (const int*)d_in[11];
    const int* i2u_row = (const int*)d_in[12];
    const int* i2u_col = (const int*)d_in[13];
    float* out = (float*)d_out;   // [u_next (M x 128) | i_next (N x 128)]

    // Workspace carve-up (~82 MB of fp32)
    float* ws = (float*)d_ws;
    float* u_f     = ws; ws += (size_t)M_NODES * DIM;
    float* i_f     = ws; ws += (size_t)N_NODES * DIM;
    float* s_u_src = ws; ws += (size_t)M_NODES * 2;   // u . a_u_src[h]
    float* s_u_dst = ws; ws += (size_t)M_NODES * 2;   // u . a_i_dst[h]
    float* s_i_src = ws; ws += (size_t)N_NODES * 2;   // i . a_i_src[h]
    float* s_i_dst = ws; ws += (size_t)N_NODES * 2;   // i . a_u_dst[h]
    float* mx_u    = ws; ws += (size_t)M_NODES * 2;
    float* sum_u   = ws; ws += (size_t)M_NODES * 2;
    float* mx_i    = ws; ws += (size_t)N_NODES * 2;
    float* sum_i   = ws; ws += (size_t)N_NODES * 2;

    const float NEG_INF = -__builtin_inff();
    const size_t outN = (size_t)(M_NODES + N_NODES) * DIM;

    // Re-init accumulators every call (deterministic across graph replays)
    fill_kernel<<<4096, 256, 0, stream>>>(out, 0.0f, outN);
    fill_kernel<<<256, 256, 0, stream>>>(mx_u, NEG_INF, (size_t)M_NODES * 2);
    fill_kernel<<<256, 256, 0, stream>>>(sum_u, 0.0f,   (size_t)M_NODES * 2);
    fill_kernel<<<128, 256, 0, stream>>>(mx_i, NEG_INF, (size_t)N_NODES * 2);
    fill_kernel<<<128, 256, 0, stream>>>(sum_i, 0.0f,   (size_t)N_NODES * 2);

    // Dense projections on the fp32 matrix pipe
    gemm_wmma_f32<<<(M_NODES / 16 + 7) / 8, 256, 0, stream>>>(u_prev, w_user, b_user, u_f, M_NODES);
    gemm_wmma_f32<<<(N_NODES / 16 + 7) / 8, 256, 0, stream>>>(i_prev, w_item, b_item, i_f, N_NODES);

    // Per-node attention scalars (kills the E x D logit gathers)
    node_scores<<<(M_NODES + 7) / 8, 256, 0, stream>>>(u_f, M_NODES, a_u_src, a_i_dst, s_u_src, s_u_dst);
    node_scores<<<(N_NODES + 7) / 8, 256, 0, stream>>>(i_f, N_NODES, a_i_src, a_u_dst, s_i_src, s_i_dst);

    const int eb = (NEDGE + 255) / 256;       // 1 thread / edge
    const int ebw = (NEDGE + 7) / 8;          // 1 wave  / edge

    // Direction u2i: segments over users (M), neighbors are items
    edge_max   <<<eb,  256, 0, stream>>>(u2i_row, u2i_col, s_u_src, s_i_dst, mx_u, NEDGE);
    edge_expsum<<<eb,  256, 0, stream>>>(u2i_row, u2i_col, s_u_src, s_i_dst, mx_u, sum_u, NEDGE);
    edge_scatter<<<ebw, 256, 0, stream>>>(u2i_row, u2i_col, s_u_src, s_i_dst, mx_u, sum_u,
                                          i_f, out, NEDGE);

    // Direction i2u: segments over items (N), neighbors are users
    edge_max   <<<eb,  256, 0, stream>>>(i2u_row, i2u_col, s_i_src, s_u_dst, mx_i, NEDGE);
    edge_expsum<<<eb,  256, 0, stream>>>(i2u_row, i2u_col, s_i_src, s_u_dst, mx_i, sum_i, NEDGE);
    edge_scatter<<<ebw, 256, 0, stream>>>(i2u_row, i2u_col, s_i_src, s_u_dst, mx_i, sum_i,
                                          u_f, out + (size_t)M_NODES * DIM, NEDGE);

    elu_kernel<<<4096, 256, 0, stream>>>(out, outN);
}